// ProposalLoss_42563125903952
// MI455X (gfx1250) — compile-verified
//
#include <hip/hip_runtime.h>
#include <math.h>

typedef __attribute__((ext_vector_type(2))) float v2f;
typedef __attribute__((ext_vector_type(8))) float v8f;
typedef __attribute__((ext_vector_type(4))) int v4i;

#define B_ 4
#define N_ 2048
#define G_ 16

// CDNA5 async global->LDS path (guarded; falls back to plain LDS copy)
#if defined(__has_builtin)
#if __has_builtin(__builtin_amdgcn_global_load_async_to_lds_b128)
#define HAVE_ASYNC_LDS 1
#endif
#if __has_builtin(__builtin_amdgcn_s_wait_asynccnt)
#define HAVE_WAIT_ASYNC 1
#endif
#endif

#ifdef HAVE_ASYNC_LDS
typedef __attribute__((address_space(1))) v4i as1_v4i;  // global
typedef __attribute__((address_space(3))) v4i as3_v4i;  // LDS
__device__ __forceinline__ void async_cp16(const void* g, void* l) {
  __builtin_amdgcn_global_load_async_to_lds_b128(
      (as1_v4i*)(void*)(unsigned long long)g, (as3_v4i*)l, 0, 0);
}
__device__ __forceinline__ void async_wait0() {
#ifdef HAVE_WAIT_ASYNC
  __builtin_amdgcn_s_wait_asynccnt(0);
#else
  asm volatile("s_wait_asynccnt 0" ::: "memory");
#endif
}
#endif

// ---------------------------------------------------------------------------
// K1: per-point precompute: sq = |p|^2, labbits = 16-bit group mask,
//     labmax = (any group) * (1 - viewed), per-batch: active mask, sum(labmax),
//     max(proposal)   (for numerically-stable log_softmax later)
// ---------------------------------------------------------------------------
__global__ void prep_kernel(const float* __restrict__ mask_output,
                            const float* __restrict__ points,
                            const float* __restrict__ lab,
                            const float* __restrict__ viewed,
                            float* __restrict__ sq,
                            float* __restrict__ labmax,
                            unsigned* __restrict__ labbits,
                            float* __restrict__ activef,
                            float* __restrict__ labsum,
                            float* __restrict__ pmax) {
  const int b = blockIdx.x;
  const int tid = threadIdx.x;
  __shared__ unsigned s_or[256];
  __shared__ float s_sum[256];
  __shared__ float s_max[256];

  const float* P = points + (size_t)b * 3 * N_;
  const float* L = lab + (size_t)b * G_ * N_;
  const float* PR = mask_output + (size_t)b * 2 * N_;

  unsigned my_or = 0u;
  float my_sum = 0.f;
  float my_max = -__builtin_inff();

  for (int n = tid; n < N_; n += 256) {
    float p0 = P[n], p1 = P[N_ + n], p2 = P[2 * N_ + n];
    sq[b * N_ + n] = p0 * p0 + p1 * p1 + p2 * p2;
    unsigned bits = 0u;
#pragma unroll
    for (int g = 0; g < G_; ++g)
      if (L[g * N_ + n] != 0.f) bits |= (1u << g);
    labbits[b * N_ + n] = bits;
    my_or |= bits;
    float vw = (viewed[b * N_ + n] > 0.f) ? 1.f : 0.f;
    float lm = (bits != 0u) ? (1.f - vw) : 0.f;  // max_g lab*(1-viewed)
    labmax[b * N_ + n] = lm;
    my_sum += lm;
    my_max = fmaxf(my_max, PR[n]);
  }
  s_or[tid] = my_or; s_sum[tid] = my_sum; s_max[tid] = my_max;
  __syncthreads();
  for (int s = 128; s > 0; s >>= 1) {
    if (tid < s) {
      s_or[tid] |= s_or[tid + s];
      s_sum[tid] += s_sum[tid + s];
      s_max[tid] = fmaxf(s_max[tid], s_max[tid + s]);
    }
    __syncthreads();
  }
  if (tid < G_) activef[b * G_ + tid] = ((s_or[0] >> tid) & 1u) ? 1.f : 0.f;
  if (tid == 0) { labsum[b] = s_sum[0]; pmax[b] = s_max[0]; }
}

// ---------------------------------------------------------------------------
// K2: fused pairwise-distance + masked group max + group min.
//     Grid (N/16, B), block = 128 (4 waves). Each block owns one 16-row i-tile;
//     each wave owns 4 groups and loops over all 128 j-tiles.
//     Per-batch row data staged into LDS once (async-to-LDS on CDNA5);
//     hot loop = ds_load + WMMA + u32-ordered masked max (no canonicalize).
// ---------------------------------------------------------------------------
__global__ void __launch_bounds__(128) distmax_kernel(
    const float* __restrict__ points,
    const float* __restrict__ sq,
    const unsigned* __restrict__ labbits,
    const float* __restrict__ activef,
    float* __restrict__ distmin) {
  const int itile = blockIdx.x;
  const int b = blockIdx.y;
  const int tid = threadIdx.x;
  const int wave = tid >> 5;
  const int lane = tid & 31;
  const int half = lane >> 4;  // K pair select for A/B, row-half for C/D
  const int l = lane & 15;     // M index (A) / N index (B, C/D)
  const int i0 = itile * 16;

  const float* P = points + (size_t)b * 3 * N_;
  const float* SQ = sq + (size_t)b * N_;
  const unsigned* LB = labbits + (size_t)b * N_;

  // LDS staging of the whole batch row: 5 * 2048 * 4B = 40KB (of 320KB/WGP)
  __shared__ float s_p0[N_];
  __shared__ float s_p1[N_];
  __shared__ float s_p2[N_];
  __shared__ float s_sq[N_];
  __shared__ unsigned s_bits[N_];

#ifdef HAVE_ASYNC_LDS
  for (int n = tid * 4; n < N_; n += 128 * 4) {
    async_cp16(P + n,          &s_p0[n]);
    async_cp16(P + N_ + n,     &s_p1[n]);
    async_cp16(P + 2 * N_ + n, &s_p2[n]);
    async_cp16(SQ + n,         &s_sq[n]);
    async_cp16(LB + n,         &s_bits[n]);
  }
  async_wait0();
  __syncthreads();
#else
  for (int n = tid; n < N_; n += 128) {
    s_p0[n] = P[n];
    s_p1[n] = P[N_ + n];
    s_p2[n] = P[2 * N_ + n];
    s_sq[n] = SQ[n];
    s_bits[n] = LB[n];
  }
  __syncthreads();
#endif

  // A tile 16x4 (MxK): lanes 0-15 hold K={0,1}, lanes 16-31 hold K={2,3(=0 pad)}
  v2f a;
  a.x = half ? s_p2[i0 + l] : s_p0[i0 + l];
  a.y = half ? 0.f : s_p1[i0 + l];

  // |p_i|^2 for the 8 rows this lane-half owns in the C/D layout
  float sqi[8];
#pragma unroll
  for (int r = 0; r < 8; ++r) sqi[r] = s_sq[i0 + r + 8 * half];

  const int g0 = wave * 4;  // this wave's 4 groups
  float af[4];
#pragma unroll
  for (int gg = 0; gg < 4; ++gg) af[gg] = activef[b * G_ + g0 + gg];

  // u32-ordered accumulators: all candidates are >= 0.0f, so float order ==
  // unsigned order on the bit pattern; avoids per-iteration canonicalization.
  unsigned macc[4][8];
#pragma unroll
  for (int gg = 0; gg < 4; ++gg)
#pragma unroll
    for (int r = 0; r < 8; ++r) macc[gg][r] = 0u;  // bits of +0.0f

#pragma unroll 2
  for (int jt = 0; jt < N_ / 16; ++jt) {
    const int j0 = jt * 16;
    // B tile 4x16 (KxN): same striping as A (gram matrix is symmetric)
    v2f bv;
    bv.x = half ? s_p2[j0 + l] : s_p0[j0 + l];
    bv.y = half ? 0.f : s_p1[j0 + l];
    float sqj = s_sq[j0 + l];
    unsigned bits = s_bits[j0 + l];

    v8f c = {};
    // inner[m][n] = sum_c p[c][i0+m] * p[c][j0+n]
    v8f D = __builtin_amdgcn_wmma_f32_16x16x4_f32(
        false, a, false, bv, (short)0, c, false, false);

#pragma unroll
    for (int r = 0; r < 8; ++r) {
      float d = fmaf(D[r], -2.0f, sqi[r] + sqj);  // squared distance
      unsigned du = __float_as_uint(fmaxf(d, 0.0f));  // clamp: candidate >= 0
#pragma unroll
      for (int gg = 0; gg < 4; ++gg) {
        // candidate = d * lab (lab in {0,1}) -> select, then unsigned max
        unsigned cand = ((bits >> (g0 + gg)) & 1u) ? du : 0u;
        macc[gg][r] = macc[gg][r] > cand ? macc[gg][r] : cand;
      }
    }
  }

  // finish row max: reduce across the 16 lanes of each half (wave32-safe)
#pragma unroll
  for (int gg = 0; gg < 4; ++gg)
#pragma unroll
    for (int r = 0; r < 8; ++r) {
      unsigned v = macc[gg][r];
#pragma unroll
      for (int off = 1; off < 16; off <<= 1) {
        unsigned o = (unsigned)__shfl_xor((int)v, off, 32);
        v = v > o ? v : o;
      }
      macc[gg][r] = v;
    }

  // scale by active and take min over this wave's 4 groups
  float minv[8];
#pragma unroll
  for (int r = 0; r < 8; ++r) {
    float mv = __builtin_inff();
#pragma unroll
    for (int gg = 0; gg < 4; ++gg) {
      float dg = (__uint_as_float(macc[gg][r]) + 1e-4f) / (af[gg] + 1e-8f);
      mv = fminf(mv, dg);
    }
    minv[r] = mv;
  }

  __shared__ float lds_min[4][16];
  if (l == 0) {
#pragma unroll
    for (int r = 0; r < 8; ++r) lds_min[wave][r + 8 * half] = minv[r];
  }
  __syncthreads();
  if (tid < 16) {
    float mv = lds_min[0][tid];
    mv = fminf(mv, lds_min[1][tid]);
    mv = fminf(mv, lds_min[2][tid]);
    mv = fminf(mv, lds_min[3][tid]);
    distmin[(size_t)b * N_ + i0 + tid] = mv;
  }
}

// ---------------------------------------------------------------------------
// K3: per-batch reductions: logsumexp, radius loss, proposal loss, conf term
// ---------------------------------------------------------------------------
__global__ void batchred_kernel(const float* __restrict__ mask_output,
                                const float* __restrict__ global_output,
                                const float* __restrict__ distmin,
                                const float* __restrict__ labmax,
                                const float* __restrict__ labsum,
                                const float* __restrict__ pmax,
                                float* __restrict__ prop,
                                float* __restrict__ rad,
                                float* __restrict__ conf) {
  const int b = blockIdx.x;
  const int tid = threadIdx.x;
  __shared__ float s_a[256], s_b[256];
  const float* pr = mask_output + (size_t)b * 2 * N_;  // proposal row
  const float* rm = pr + N_;                           // radius row
  const float pm = pmax[b];

  float sexp = 0.f, srad = 0.f;
  for (int n = tid; n < N_; n += 256) {
    sexp += expf(pr[n] - pm);
    float diff = rm[n] - distmin[b * N_ + n];
    srad += diff * diff * labmax[b * N_ + n];
  }
  s_a[tid] = sexp; s_b[tid] = srad;
  __syncthreads();
  for (int s = 128; s > 0; s >>= 1) {
    if (tid < s) { s_a[tid] += s_a[tid + s]; s_b[tid] += s_b[tid + s]; }
    __syncthreads();
  }
  const float lse = logf(s_a[0]);
  const float radtot = s_b[0];
  const float denom = labsum[b] + 1e-8f;
  __syncthreads();

  float sprop = 0.f;
  for (int n = tid; n < N_; n += 256) {
    float lnorm = labmax[b * N_ + n] / denom;
    sprop += -lnorm * (pr[n] - pm - lse);  // -lab_norm * log_softmax
  }
  s_a[tid] = sprop;
  __syncthreads();
  for (int s = 128; s > 0; s >>= 1) {
    if (tid < s) s_a[tid] += s_a[tid + s];
    __syncthreads();
  }
  if (tid == 0) {
    prop[b] = s_a[0];
    rad[b] = radtot;
    float maxlab = (labsum[b] > 0.f) ? 1.f : 0.f;  // labmax in {0,1}
    float f = maxlab / denom;                      // finish_label
    float y = f / (f + 1e-12f);
    float z = global_output[b];
    conf[b] = fmaxf(z, 0.f) - z * y + log1pf(expf(-fabsf(z)));
  }
}

// ---------------------------------------------------------------------------
// K4: combine the B per-batch partials into the 3 scalar outputs
// ---------------------------------------------------------------------------
__global__ void final_kernel(const float* __restrict__ prop,
                             const float* __restrict__ rad,
                             const float* __restrict__ conf,
                             float* __restrict__ out) {
  if (threadIdx.x == 0 && blockIdx.x == 0) {
    float ps = 0.f, rs = 0.f, cs = 0.f;
    for (int b = 0; b < B_; ++b) { ps += prop[b]; rs += rad[b]; cs += conf[b]; }
    out[0] = ps / (float)B_;
    out[1] = rs / (float)B_;
    out[2] = cs / (float)B_;
  }
}

extern "C" void kernel_launch(void* const* d_in, const int* in_sizes, int n_in,
                              void* d_out, int out_size, void* d_ws, size_t ws_size,
                              hipStream_t stream) {
  const float* mask_output   = (const float*)d_in[0];  // (B,2,N)
  const float* global_output = (const float*)d_in[1];  // (B,1)
  const float* points        = (const float*)d_in[2];  // (B,3,N)
  const float* lab           = (const float*)d_in[3];  // (B,G,N)
  const float* viewed        = (const float*)d_in[4];  // (B,1,N)
  float* out = (float*)d_out;                          // 3 scalars

  // workspace carve-up (~132 KB)
  float* f        = (float*)d_ws;
  float* sq       = f;                      // B*N
  float* labmax   = sq + B_ * N_;           // B*N
  float* distmin  = labmax + B_ * N_;       // B*N
  float* activef  = distmin + B_ * N_;      // B*G
  float* labsum   = activef + B_ * G_;      // B
  float* pmaxp    = labsum + B_;            // B
  float* prop     = pmaxp + B_;             // B
  float* rad      = prop + B_;              // B
  float* conf     = rad + B_;               // B
  unsigned* labbits = (unsigned*)(conf + B_); // B*N u32

  prep_kernel<<<dim3(B_), dim3(256), 0, stream>>>(
      mask_output, points, lab, viewed, sq, labmax, labbits, activef, labsum, pmaxp);

  distmax_kernel<<<dim3(N_ / 16, B_), dim3(128), 0, stream>>>(
      points, sq, labbits, activef, distmin);

  batchred_kernel<<<dim3(B_), dim3(256), 0, stream>>>(
      mask_output, global_output, distmin, labmax, labsum, pmaxp, prop, rad, conf);

  final_kernel<<<1, 32, 0, stream>>>(prop, rad, conf, out);
}